// SingleHeadGeneralGATLayer_42245298323498
// MI455X (gfx1250) — compile-verified
//
#include <hip/hip_runtime.h>
#include <hip/hip_bf16.h>
#include <math.h>

typedef __attribute__((ext_vector_type(2))) float v2f;
typedef __attribute__((ext_vector_type(8))) float v8f;

#define GAT_ALPHA 0.2f
#define GB 8
#define GN 2048
#define GDIN 256
#define GDOUT 128

// ---------------------------------------------------------------------------
// Kernel 1: wh = x @ W^T   (M=B*N=16384, K=256, N=128), f32 WMMA 16x16x4.
// 8 waves per block, wave w computes n-tile w; block handles one 16-row slab.
// ---------------------------------------------------------------------------
__global__ __launch_bounds__(256)
void gat_wh_kernel(const float* __restrict__ x, const float* __restrict__ W,
                   float* __restrict__ wh) {
  const int tid  = threadIdx.x;
  const int wave = tid >> 5;
  const int lane = tid & 31;
  const int li   = lane & 15;

  const int m_base = blockIdx.x * 16;        // row slab in [0, 16384)
  const int n_base = wave * 16;              // 8 waves * 16 = 128 cols

  const int koff = (lane < 16) ? 0 : 2;      // A/B fragment K split

  const float* __restrict__ xrow = x + (size_t)(m_base + li) * GDIN;
  const float* __restrict__ Wrow = W + (size_t)(n_base + li) * GDIN; // B[k][n]=W[n][k]

  v8f c = {};
#pragma unroll 4
  for (int k0 = 0; k0 < GDIN; k0 += 4) {
    v2f a = { xrow[k0 + koff], xrow[k0 + koff + 1] };
    v2f b = { Wrow[k0 + koff], Wrow[k0 + koff + 1] };
    c = __builtin_amdgcn_wmma_f32_16x16x4_f32(false, a, false, b,
                                              (short)0, c, false, false);
  }

  const int nn   = n_base + li;
  const int mtop = m_base + ((lane < 16) ? 0 : 8);
#pragma unroll
  for (int r = 0; r < 8; ++r)
    wh[(size_t)(mtop + r) * GDOUT + nn] = c[r];
}

// ---------------------------------------------------------------------------
// Kernel 2: s[row] = wh[row]·a_src, d[row] = wh[row]·a_dst  (wave per row)
// ---------------------------------------------------------------------------
__global__ __launch_bounds__(256)
void gat_score_kernel(const float* __restrict__ wh, const float* __restrict__ a,
                      float* __restrict__ s, float* __restrict__ d) {
  const int tid  = threadIdx.x;
  const int wave = tid >> 5;
  const int lane = tid & 31;
  const int row  = blockIdx.x * 8 + wave;    // 2048 blocks * 8 waves = 16384

  const float* __restrict__ whr = wh + (size_t)row * GDOUT;
  float ssum = 0.f, dsum = 0.f;
#pragma unroll
  for (int o = lane; o < GDOUT; o += 32) {
    float v = whr[o];
    ssum += v * a[o];
    dsum += v * a[GDOUT + o];
  }
#pragma unroll
  for (int off = 16; off > 0; off >>= 1) {
    ssum += __shfl_down(ssum, off, 32);
    dsum += __shfl_down(dsum, off, 32);
  }
  if (lane == 0) { s[row] = ssum; d[row] = dsum; }
}

// ---------------------------------------------------------------------------
// Kernel 3: per-row online softmax stats (max m, sum l) over masked
// leakyrelu(s_i + d_j), + pack adj row into a 2048-bit mask (reads the 134MB
// adj exactly once for the whole pipeline). One 256-thread block per row.
// ---------------------------------------------------------------------------
__global__ __launch_bounds__(256)
void gat_rowstat_kernel(const int* __restrict__ adj,
                        const float* __restrict__ s, const float* __restrict__ d,
                        float* __restrict__ mrow, float* __restrict__ lrow,
                        unsigned* __restrict__ amask) {
  const int tid  = threadIdx.x;
  const int wave = tid >> 5;
  const int lane = tid & 31;
  const int row  = blockIdx.x;               // 16384 rows
  const int b    = row >> 11;

  const int*   __restrict__ adjr = adj + (size_t)row * GN;
  const float* __restrict__ db   = d + (size_t)b * GN;
  unsigned*    __restrict__ mkr  = amask + (size_t)row * (GN / 32);
  const float si = s[row];

  float m = -INFINITY, l = 0.f;
#pragma unroll
  for (int k = 0; k < GN / 256; ++k) {
    const int j  = k * 256 + tid;
    const int ad = adjr[j];
    unsigned bal = (unsigned)__ballot(ad != 0);
    if (lane == 0) mkr[k * 8 + wave] = bal;   // bits j = k*256+wave*32+lane
    if (ad) {
      float e = si + db[j];
      e = (e >= 0.f) ? e : GAT_ALPHA * e;
      if (e > m) { l = l * __expf(m - e) + 1.f; m = e; }
      else       { l += __expf(e - m); }
    }
  }

  __shared__ float sm[256], sl[256];
  sm[tid] = m; sl[tid] = l;
  __syncthreads();
  for (int off = 128; off > 0; off >>= 1) {
    if (tid < off) {
      float m1 = sm[tid], l1 = sl[tid];
      float m2 = sm[tid + off], l2 = sl[tid + off];
      float M = fmaxf(m1, m2);
      float L = 0.f;
      if (m1 != -INFINITY) L += l1 * __expf(m1 - M);
      if (m2 != -INFINITY) L += l2 * __expf(m2 - M);
      sm[tid] = M; sl[tid] = L;
    }
    __syncthreads();
  }
  if (tid == 0) { mrow[row] = sm[0]; lrow[row] = sl[0]; }
}

// ---------------------------------------------------------------------------
// Kernel 4: out = relu(softmax(e) @ wh), fused. Attention A-fragments are
// rebuilt on the fly in WMMA layout from the bitmask + (s,d,m,l); each wave
// carries 4 accumulator tiles (64 output cols) so exp work is amortized.
// Block: 64 threads (2 waves) covers 16 rows x 128 cols.
// ---------------------------------------------------------------------------
__global__ __launch_bounds__(64)
void gat_out_kernel(const unsigned* __restrict__ amask,
                    const float* __restrict__ wh,
                    const float* __restrict__ s, const float* __restrict__ d,
                    const float* __restrict__ mrow, const float* __restrict__ lrow,
                    float* __restrict__ out) {
  constexpr int NT = 4;
  const int tid   = threadIdx.x;
  const int wave  = tid >> 5;                // 0..1
  const int lane  = tid & 31;
  const int li    = lane & 15;
  const int b     = blockIdx.x >> 7;         // 8 * 128 = 1024 blocks
  const int itile = blockIdx.x & 127;

  const int grow = b * GN + itile * 16 + li; // global row for this lane's A-row
  const float si   = s[grow];
  const float mi   = mrow[grow];
  const float lv   = lrow[grow];
  const float linv = (lv > 0.f) ? (1.f / lv) : 0.f;
  const int   koff = (lane < 16) ? 0 : 2;

  const unsigned* __restrict__ mkr = amask + (size_t)grow * (GN / 32);
  const float*    __restrict__ db  = d + (size_t)b * GN;
  const float*    __restrict__ whb = wh + (size_t)b * GN * GDOUT;
  const int nb0 = wave * NT * 16;

  v8f c[NT] = {};
  for (int j0 = 0; j0 < GN; j0 += 4) {
    const unsigned mw = mkr[j0 >> 5];
    const int j = j0 + koff;
    float e0 = si + db[j];     e0 = (e0 >= 0.f) ? e0 : GAT_ALPHA * e0;
    float e1 = si + db[j + 1]; e1 = (e1 >= 0.f) ? e1 : GAT_ALPHA * e1;
    float p0 = ((mw >> (j & 31)) & 1u)       ? __expf(e0 - mi) * linv : 0.f;
    float p1 = ((mw >> ((j + 1) & 31)) & 1u) ? __expf(e1 - mi) * linv : 0.f;
    v2f a = { p0, p1 };
    const float* __restrict__ wj = whb + (size_t)j * GDOUT; // K rows j, j+1
#pragma unroll
    for (int t = 0; t < NT; ++t) {
      const int nn = nb0 + t * 16 + li;
      v2f bv = { wj[nn], wj[GDOUT + nn] };
      c[t] = __builtin_amdgcn_wmma_f32_16x16x4_f32(false, a, false, bv,
                                                   (short)0, c[t], false, false);
    }
  }

  const int mtop = itile * 16 + ((lane < 16) ? 0 : 8);
#pragma unroll
  for (int t = 0; t < NT; ++t) {
    const int nn = nb0 + t * 16 + li;
#pragma unroll
    for (int r = 0; r < 8; ++r) {
      float v = c[t][r];
      out[((size_t)(b * GN + mtop + r)) * GDOUT + nn] = (v > 0.f) ? v : 0.f;
    }
  }
}

// ---------------------------------------------------------------------------
extern "C" void kernel_launch(void* const* d_in, const int* in_sizes, int n_in,
                              void* d_out, int out_size, void* d_ws, size_t ws_size,
                              hipStream_t stream) {
  const float* x   = (const float*)d_in[0];  // (8,2048,256)
  const int*   adj = (const int*)d_in[1];    // (8,2048,2048)
  const float* W   = (const float*)d_in[2];  // (128,256)
  const float* a   = (const float*)d_in[3];  // (256,)
  float* out = (float*)d_out;                // (8,2048,128)

  // workspace layout
  float* wh   = (float*)d_ws;                       // 16384*128 f32 = 8 MB
  float* s    = wh + (size_t)GB * GN * GDOUT;       // 16384
  float* dv   = s + GB * GN;                        // 16384
  float* mrow = dv + GB * GN;                       // 16384
  float* lrow = mrow + GB * GN;                     // 16384
  unsigned* amask = (unsigned*)(lrow + GB * GN);    // 16384*64 words = 4 MB

  gat_wh_kernel<<<(GB * GN) / 16, 256, 0, stream>>>(x, W, wh);
  gat_score_kernel<<<(GB * GN) / 8, 256, 0, stream>>>(wh, a, s, dv);
  gat_rowstat_kernel<<<GB * GN, 256, 0, stream>>>(adj, s, dv, mrow, lrow, amask);
  gat_out_kernel<<<GB * (GN / 16), 64, 0, stream>>>(amask, wh, s, dv, mrow, lrow, out);
}